// HyperGraphConvolution_29978871726195
// MI455X (gfx1250) — compile-verified
//
#include <hip/hip_runtime.h>
#include <hip/hip_bf16.h>

// ---------------------------------------------------------------------------
// HyperGraphConvolution for MI455X (gfx1250, wave32)
//   HW = H @ W                     : f32 WMMA 16x16x4 (exact f32 path)
//   p  = (HW @ emb_w + emb_b) * rv : wave-per-node reduction
//   out = HW + bias                : vectorized init
//   scatter: out[u]+=HW[c1], out[v]+=HW[c2] via global f32 atomics (L2-resident)
// ---------------------------------------------------------------------------

typedef float v2f __attribute__((ext_vector_type(2)));
typedef float v8f __attribute__((ext_vector_type(8)));

#define A_DIM 512   // inner dim (H columns)
#define B_DIM 256   // output feature dim

// ---------------------------------------------------------------------------
// GEMM: HW[n,256] = H[n,512] @ W[512,256] with V_WMMA_F32_16X16X4_F32.
// Block = 128 threads (4 waves). Block covers 16 rows x 256 cols;
// each wave computes a 16x64 tile (4 accumulators), K-loop step 4.
// ---------------------------------------------------------------------------
__global__ __launch_bounds__(128)
void gemm_hw_wmma(const float* __restrict__ H, const float* __restrict__ W,
                  float* __restrict__ HW, int n) {
  const int m0   = blockIdx.x * 16;
  const int wave = threadIdx.x >> 5;    // 0..3
  const int lane = threadIdx.x & 31;
  const int half = lane >> 4;           // 0: K 0..1, 1: K 2..3
  const int l16  = lane & 15;
  const int n0   = wave * 64;

  // clamp A row for (never-taken at n=10000) tail safety
  const int arow = (m0 + l16 < n) ? (m0 + l16) : (n - 1);
  const float* __restrict__ aPtr = H + (size_t)arow * A_DIM + 2 * half;

  v8f acc0 = {}, acc1 = {}, acc2 = {}, acc3 = {};

  for (int k0 = 0; k0 < A_DIM; k0 += 4) {
    // A fragment: 16x4, lane holds A[row][k0+2*half .. +1]
    v2f a;
    a.x = aPtr[k0];
    a.y = aPtr[k0 + 1];

    // B fragments: 4x16 each; lane holds B[k0+2*half + {0,1}][n0+16t+l16]
    const float* __restrict__ bPtr = W + (size_t)(k0 + 2 * half) * B_DIM + l16;
    v2f b0, b1, b2, b3;
    b0.x = bPtr[n0 +  0]; b0.y = bPtr[B_DIM + n0 +  0];
    b1.x = bPtr[n0 + 16]; b1.y = bPtr[B_DIM + n0 + 16];
    b2.x = bPtr[n0 + 32]; b2.y = bPtr[B_DIM + n0 + 32];
    b3.x = bPtr[n0 + 48]; b3.y = bPtr[B_DIM + n0 + 48];

    acc0 = __builtin_amdgcn_wmma_f32_16x16x4_f32(false, a, false, b0, (short)0, acc0, false, false);
    acc1 = __builtin_amdgcn_wmma_f32_16x16x4_f32(false, a, false, b1, (short)0, acc1, false, false);
    acc2 = __builtin_amdgcn_wmma_f32_16x16x4_f32(false, a, false, b2, (short)0, acc2, false, false);
    acc3 = __builtin_amdgcn_wmma_f32_16x16x4_f32(false, a, false, b3, (short)0, acc3, false, false);
  }

  // D layout: VGPR r holds row (m0 + half*8 + r), col = n0 + 16t + l16
  #pragma unroll
  for (int r = 0; r < 8; ++r) {
    const int row = m0 + half * 8 + r;
    if (row < n) {
      float* __restrict__ o = HW + (size_t)row * B_DIM + l16;
      o[n0 +  0] = acc0[r];
      o[n0 + 16] = acc1[r];
      o[n0 + 32] = acc2[r];
      o[n0 + 48] = acc3[r];
    }
  }
}

// ---------------------------------------------------------------------------
// p[i] = (dot(HW[i,:], emb_w) + emb_b) * rv     (one wave per node)
// ---------------------------------------------------------------------------
__global__ __launch_bounds__(256)
void compute_p(const float* __restrict__ HW, const float* __restrict__ emb_w,
               const float* __restrict__ emb_b, const float* __restrict__ rv,
               float* __restrict__ p, int n) {
  const int node = blockIdx.x * 8 + (threadIdx.x >> 5);
  const int lane = threadIdx.x & 31;
  if (node >= n) return;
  const float* __restrict__ row = HW + (size_t)node * B_DIM;
  float s = 0.0f;
  #pragma unroll
  for (int j = 0; j < B_DIM / 32; ++j)
    s += row[lane + 32 * j] * emb_w[lane + 32 * j];
  #pragma unroll
  for (int off = 16; off > 0; off >>= 1)
    s += __shfl_xor(s, off, 32);
  if (lane == 0) p[node] = (s + emb_b[0]) * rv[0];
}

// ---------------------------------------------------------------------------
// out = HW + bias (float4-vectorized; total is a multiple of 4)
// ---------------------------------------------------------------------------
__global__ __launch_bounds__(256)
void init_out(const float* __restrict__ HW, const float* __restrict__ bias,
              float* __restrict__ out, int total4) {
  const int i = blockIdx.x * blockDim.x + threadIdx.x;
  if (i >= total4) return;
  const float4 h = ((const float4*)HW)[i];
  const int c = (i * 4) & (B_DIM - 1);
  float4 o;
  o.x = h.x + bias[c + 0];
  o.y = h.y + bias[c + 1];
  o.z = h.z + bias[c + 2];
  o.w = h.w + bias[c + 3];
  ((float4*)out)[i] = o;
}

// ---------------------------------------------------------------------------
// Edge scatter: one 256-thread block per edge. cond = p[u] > p[v];
//   out[u,:] += HW[cond ? v : u, :];  out[v,:] += HW[cond ? u : v, :]
// u,v,p loads are block-uniform (scalarized); gathers/atomics coalesced.
// All traffic stays in the 192 MB L2.
// ---------------------------------------------------------------------------
__global__ __launch_bounds__(256)
void scatter_edges(const long long* __restrict__ structure,
                   const float* __restrict__ p, const float* __restrict__ HW,
                   float* __restrict__ out, int E) {
  const int j = threadIdx.x;  // feature column 0..255
  for (int e = blockIdx.x; e < E; e += gridDim.x) {
    const int u = (int)structure[e];
    const int v = (int)structure[(size_t)E + e];
    const bool cond = p[u] > p[v];
    const int c1 = cond ? v : u;
    const int c2 = cond ? u : v;
    atomicAdd(&out[(size_t)u * B_DIM + j], HW[(size_t)c1 * B_DIM + j]);
    atomicAdd(&out[(size_t)v * B_DIM + j], HW[(size_t)c2 * B_DIM + j]);
  }
}

// ---------------------------------------------------------------------------
extern "C" void kernel_launch(void* const* d_in, const int* in_sizes, int n_in,
                              void* d_out, int out_size, void* d_ws, size_t ws_size,
                              hipStream_t stream) {
  const long long* structure = (const long long*)d_in[0];  // (2, E) int64
  const float*     H         = (const float*)d_in[1];      // (n, 512)
  const float*     W         = (const float*)d_in[2];      // (512, 256)
  const float*     bias      = (const float*)d_in[3];      // (256,)
  const float*     emb_w     = (const float*)d_in[4];      // (256, 1)
  const float*     emb_b     = (const float*)d_in[5];      // (1,)
  const float*     rv        = (const float*)d_in[6];      // (1,)
  float*           out       = (float*)d_out;              // (n, 256)

  const int n = out_size / B_DIM;      // 10000
  const int E = in_sizes[0] / 2;       // 320000

  float* HW = (float*)d_ws;                      // n*256 floats (10.24 MB)
  float* p  = HW + (size_t)n * B_DIM;            // n floats

  // 1) GEMM via f32 WMMA: 625 blocks x 128 threads
  gemm_hw_wmma<<<(n + 15) / 16, 128, 0, stream>>>(H, W, HW, n);

  // 2) projection p
  compute_p<<<(n + 7) / 8, 256, 0, stream>>>(HW, emb_w, emb_b, rv, p, n);

  // 3) out = HW + bias
  const int total4 = (n * B_DIM) / 4;
  init_out<<<(total4 + 255) / 256, 256, 0, stream>>>(HW, bias, out, total4);

  // 4) edge scatter with f32 atomics (L2-resident)
  scatter_edges<<<E, 256, 0, stream>>>(structure, p, HW, out, E);
}